// XuHawkesTorch_8847632629794
// MI455X (gfx1250) — compile-verified
//
#include <hip/hip_runtime.h>
#include <hip/hip_bf16.h>
#include <math.h>

// ---------------- problem constants (match reference) ----------------
#define D_DIM   1024
#define M_EV    32768
#define BLK     128                 // events per scan block
#define NBLK    (M_EV / BLK)        // 256 blocks
#define EPS_MU  1e-6f
#define EPS_LOG 1e-8f

typedef __attribute__((ext_vector_type(2))) float v2f;
typedef __attribute__((ext_vector_type(8))) float v8f;

__device__ __forceinline__ float softplus_f(float x) {
    // numerically stable softplus
    return fmaxf(x, 0.0f) + log1pf(expf(-fabsf(x)));
}

// ---------------------------------------------------------------------
// Kernel 1: init scratch — mu = softplus(log_mu)+eps, zero accumulators
// grid: 4 x 256
// ---------------------------------------------------------------------
__global__ void k_init(const float* __restrict__ log_mu,
                       float* __restrict__ mu,
                       float* __restrict__ contrib,
                       float* __restrict__ colsum,
                       float* __restrict__ llsum) {
    int d = blockIdx.x * blockDim.x + threadIdx.x;
    if (d < D_DIM) {
        mu[d]      = softplus_f(log_mu[d]) + EPS_MU;
        contrib[d] = 0.0f;
        colsum[d]  = 0.0f;
    }
    if (d == 0) llsum[0] = 0.0f;
}

// ---------------------------------------------------------------------
// Kernel 2: alpha = softplus(log_alpha) materialized once (4 MB, L2-hot),
// plus column sums for the compensator.  grid: D_DIM x 256
// ---------------------------------------------------------------------
__global__ void k_alpha(const float* __restrict__ log_alpha,
                        float* __restrict__ alpha,
                        float* __restrict__ colsum) {
    int row = blockIdx.x;
    const float* src = log_alpha + (size_t)row * D_DIM;
    float*       dst = alpha     + (size_t)row * D_DIM;
    for (int c = threadIdx.x; c < D_DIM; c += blockDim.x) {
        float a = softplus_f(src[c]);
        dst[c] = a;
        atomicAdd(&colsum[c], a);
    }
}

// ---------------------------------------------------------------------
// Kernel 3: compensator segment-sum  contrib[mark] += 1 - exp(-(T - t))
// grid: M/256 x 256
// ---------------------------------------------------------------------
__global__ void k_comp(const float* __restrict__ t,
                       const int*   __restrict__ marks,
                       const int*   __restrict__ Tmax_p,
                       float* __restrict__ contrib) {
    int i = blockIdx.x * blockDim.x + threadIdx.x;
    if (i < M_EV) {
        float Tm = (float)(*Tmax_p);
        float v  = 1.0f - expf(-(Tm - t[i]));
        atomicAdd(&contrib[marks[i]], v);
    }
}

// ---------------------------------------------------------------------
// Kernel 4: sequential recurrence over 256 blocks computing the prefix
// excitation state at each block boundary.
// blockStates[b][d'] = sum_{j < b*BLK} exp(-(t_ref_b - t_j)) [mark_j==d']
// with t_ref_b = t[b*BLK - 1].   grid: 1 x 1024 (thread = dimension)
// ---------------------------------------------------------------------
__global__ void __launch_bounds__(1024)
k_states(const float* __restrict__ t,
         const int*   __restrict__ marks,
         float* __restrict__ blockStates) {
    __shared__ float acc[D_DIM];
    int tid = threadIdx.x;           // dimension d'
    float S = 0.0f;
    float tprev = 0.0f;
    for (int b = 0; b < NBLK; ++b) {
        blockStates[(size_t)b * D_DIM + tid] = S;   // state before block b
        float te = t[(b + 1) * BLK - 1];            // last event time of block b
        S *= expf(-(te - tprev));                   // decay to te (S==0 first iter)
        acc[tid] = 0.0f;
        __syncthreads();
        if (tid < BLK) {
            int k = b * BLK + tid;
            atomicAdd(&acc[marks[k]], expf(-(te - t[k])));  // LDS atomic scatter
        }
        __syncthreads();
        S += acc[tid];
        tprev = te;
    }
}

// ---------------------------------------------------------------------
// Kernel 5: main parallel-over-blocks likelihood kernel.
//  Phase A: cross-block term via V_WMMA_F32_16X16X4_F32.
//           wave w handles events [16w,16w+16): W = A_gathered(16xK) x S_b
//           (S_b broadcast over the 16 N-columns; column 0 is the result).
//  Phase B: within-block triangular term by direct gathered accumulation.
// grid: NBLK x 256
// ---------------------------------------------------------------------
__global__ void __launch_bounds__(256)
k_main(const float* __restrict__ t,
       const int*   __restrict__ marks,
       const float* __restrict__ alpha,
       const float* __restrict__ mu,
       const float* __restrict__ blockStates,
       float* __restrict__ llsum) {
    __shared__ float sS[D_DIM];     // prefix state for this block
    __shared__ float ev_t[BLK];
    __shared__ int   ev_m[BLK];
    __shared__ float wbuf[BLK];     // cross-term dot products
    __shared__ float red[BLK];

    int b   = blockIdx.x;
    int tid = threadIdx.x;

    for (int c = tid; c < D_DIM; c += 256)
        sS[c] = blockStates[(size_t)b * D_DIM + c];
    if (tid < BLK) {
        int i = b * BLK + tid;
        ev_t[tid] = t[i];
        ev_m[tid] = marks[i];
    }
    __syncthreads();

    // ---- Phase A: WMMA cross-block dot products (EXEC all-1, no divergence)
    int lane = tid & 31;
    int wave = tid >> 5;             // 8 waves -> 8 tiles of 16 events
    int mrow = lane & 15;            // M row inside tile
    int koff = (lane >> 4) << 1;     // lanes 0-15 -> K+0,K+1 ; lanes 16-31 -> K+2,K+3
    int arow_idx = ev_m[wave * 16 + mrow];
    const float* arow = alpha + (size_t)arow_idx * D_DIM;

    v8f accf = {};
#pragma unroll 4
    for (int k = 0; k < D_DIM; k += 4) {
        v2f af, bf;
        af[0] = arow[k + koff];
        af[1] = arow[k + koff + 1];
        bf[0] = sS[k + koff];        // broadcast S over all 16 N columns
        bf[1] = sS[k + koff + 1];
        accf = __builtin_amdgcn_wmma_f32_16x16x4_f32(
            /*neg_a=*/false, af, /*neg_b=*/false, bf,
            /*c_mod=*/(short)0, accf, /*reuse_a=*/false, /*reuse_b=*/false);
    }
    // C/D layout: lanes 0-15 hold M=v (VGPR v), lanes 16-31 hold M=v+8; N=lane%16.
    // Column N=0 lives in lane 0 (M 0..7) and lane 16 (M 8..15).
    if (lane == 0) {
#pragma unroll
        for (int v = 0; v < 8; ++v) wbuf[wave * 16 + v] = accf[v];
    }
    if (lane == 16) {
#pragma unroll
        for (int v = 0; v < 8; ++v) wbuf[wave * 16 + 8 + v] = accf[v];
    }
    __syncthreads();

    // ---- Phase B: within-block triangular term + log-likelihood
    float ll = 0.0f;
    if (tid < BLK) {
        int   di = ev_m[tid];
        float ti = ev_t[tid];
        const float* ai = alpha + (size_t)di * D_DIM;   // one 4KB row, cache-hot
        float within = 0.0f;
        for (int k = 0; k < tid; ++k)
            within += ai[ev_m[k]] * expf(-(ti - ev_t[k]));
        float cross = 0.0f;
        if (b > 0) {
            float tref = t[b * BLK - 1];
            cross = expf(-(ti - tref)) * wbuf[tid];
        }
        ll = logf(mu[di] + cross + within + EPS_LOG);
        red[tid] = ll;
    }
    __syncthreads();
    if (tid == 0) {
        float s = 0.0f;
        for (int k = 0; k < BLK; ++k) s += red[k];
        atomicAdd(llsum, s);
    }
}

// ---------------------------------------------------------------------
// Kernel 6: finalize:  out = ll_sum - ( sum(mu)*T + colsum(alpha).contrib )
// grid: 1 x 1024
// ---------------------------------------------------------------------
__global__ void __launch_bounds__(1024)
k_final(const float* __restrict__ mu,
        const float* __restrict__ colsum,
        const float* __restrict__ contrib,
        const float* __restrict__ llsum,
        const int*   __restrict__ Tmax_p,
        float* __restrict__ out) {
    __shared__ float red[D_DIM];
    int d = threadIdx.x;
    float Tm = (float)(*Tmax_p);
    red[d] = mu[d] * Tm + colsum[d] * contrib[d];
    __syncthreads();
    for (int s = D_DIM / 2; s > 0; s >>= 1) {
        if (d < s) red[d] += red[d + s];
        __syncthreads();
    }
    if (d == 0) out[0] = llsum[0] - red[0];
}

// ---------------------------------------------------------------------
extern "C" void kernel_launch(void* const* d_in, const int* in_sizes, int n_in,
                              void* d_out, int out_size, void* d_ws, size_t ws_size,
                              hipStream_t stream) {
    (void)in_sizes; (void)n_in; (void)out_size; (void)ws_size;
    const float* t_events  = (const float*)d_in[0];
    const int*   marks     = (const int*)d_in[1];
    const float* log_mu    = (const float*)d_in[2];
    const float* log_alpha = (const float*)d_in[3];
    const int*   Tmax_p    = (const int*)d_in[4];
    float*       out       = (float*)d_out;

    // workspace layout (floats)
    float* ws          = (float*)d_ws;
    float* alpha       = ws;                                 // D*D
    float* blockStates = alpha + (size_t)D_DIM * D_DIM;      // NBLK*D
    float* mu          = blockStates + (size_t)NBLK * D_DIM; // D
    float* contrib     = mu + D_DIM;                         // D
    float* colsum      = contrib + D_DIM;                    // D
    float* llsum       = colsum + D_DIM;                     // 1

    k_init  <<<dim3(D_DIM / 256), dim3(256), 0, stream>>>(log_mu, mu, contrib, colsum, llsum);
    k_alpha <<<dim3(D_DIM),       dim3(256), 0, stream>>>(log_alpha, alpha, colsum);
    k_comp  <<<dim3(M_EV / 256),  dim3(256), 0, stream>>>(t_events, marks, Tmax_p, contrib);
    k_states<<<dim3(1),           dim3(1024), 0, stream>>>(t_events, marks, blockStates);
    k_main  <<<dim3(NBLK),        dim3(256), 0, stream>>>(t_events, marks, alpha, mu, blockStates, llsum);
    k_final <<<dim3(1),           dim3(1024), 0, stream>>>(mu, colsum, contrib, llsum, Tmax_p, out);
}